// RNNDecoder_34033320853744
// MI455X (gfx1250) — compile-verified
//
#include <hip/hip_runtime.h>

// Problem dims (fixed by the reference)
#define V_ 32000
#define T_ 64
#define B_ 32
#define H_ 32
#define E_ 200
#define Q_ 8            // column-stripe blocks per t in the big GEMM
#define NCHUNK_ (V_ / 128)   // 250 chunks of 128 columns

typedef __attribute__((ext_vector_type(2))) float v2f;
typedef __attribute__((ext_vector_type(8))) float v8f;

// CDNA5 hardware tanh (trans op). v_nop guards the 1-instruction
// trans-result hazard (ISA 7.4: 1 independent op or V_NOP after trans).
__device__ __forceinline__ float tanh_hw(float x) {
    float r;
    asm volatile("v_tanh_f32 %0, %1\n\tv_nop" : "=v"(r) : "v"(x));
    return r;
}

// ---------------------------------------------------------------------------
// Kernel 1: I[t][b][h] = emb_table[y[b,t]] . Wi[h,:] + bi[h] + bh[h]
// 2048 (t,b) rows, one wave per row, lane = h. ~26 MFLOP, trivial.
// Row bases are 800-byte offsets -> 16B aligned, so float4 loads are legal.
// ---------------------------------------------------------------------------
__global__ __launch_bounds__(256) void i2h_kernel(
    const int* __restrict__ y, const float* __restrict__ emb,
    const float* __restrict__ Wi, const float* __restrict__ bi,
    const float* __restrict__ bh, float* __restrict__ Ibuf)
{
    int tid  = threadIdx.x;
    int pair = blockIdx.x * 8 + (tid >> 5);   // pair = t*B + b
    int t = pair >> 5;
    int b = pair & 31;
    int h = tid & 31;

    int tok = y[b * T_ + t];
    const float4* erow = (const float4*)(emb + (size_t)tok * E_);
    const float4* wrow = (const float4*)(Wi + h * E_);
    float acc = bi[h] + bh[h];
    #pragma unroll 2
    for (int e = 0; e < E_ / 4; ++e) {
        float4 ev = erow[e], wv = wrow[e];
        acc = fmaf(ev.x, wv.x, acc);
        acc = fmaf(ev.y, wv.y, acc);
        acc = fmaf(ev.z, wv.z, acc);
        acc = fmaf(ev.w, wv.w, acc);
    }
    Ibuf[t * (B_ * H_) + b * H_ + h] = acc;
}

// ---------------------------------------------------------------------------
// Kernel 2: sequential recurrence  h = tanh(I_t + h @ Wh^T), T=64 steps.
// One block, 4 waves; wave (mi,ni) owns the 16x16 tile (m=batch, n=h_out).
// V_WMMA_F32_16X16X4_F32, K=32 -> 8 k-steps. h lives in LDS between steps.
// Branch-free body (hw tanh) keeps EXEC all-1s around every WMMA.
// ---------------------------------------------------------------------------
__global__ __launch_bounds__(128) void recurrent_kernel(
    const float* __restrict__ enc_h, const float* __restrict__ Wh,
    const float* __restrict__ Ibuf, float* __restrict__ Hall)
{
    __shared__ float hcur[B_ * H_];
    int tid  = threadIdx.x;
    int w    = tid >> 5;
    int lane = tid & 31;
    int mi = w >> 1, ni = w & 1;
    int l15    = lane & 15;
    int khalf  = (lane < 16) ? 0 : 2;   // A/B: upper half-wave holds K+2,K+3
    int rowoff = (lane < 16) ? 0 : 8;   // C/D: upper half-wave holds M=r+8

    for (int i = tid; i < B_ * H_; i += 128) hcur[i] = enc_h[i];

    // Preload B-matrix tiles: B[k][n] = Wh[n][k], n = ni*16 + l15
    v2f bb[8];
    {
        const float* wrow = Wh + (ni * 16 + l15) * H_;
        #pragma unroll
        for (int k0 = 0; k0 < 8; ++k0) {
            bb[k0].x = wrow[4 * k0 + khalf];
            bb[k0].y = wrow[4 * k0 + khalf + 1];
        }
    }
    __syncthreads();

    for (int t = 0; t < T_; ++t) {
        // A tile from LDS: A[m][k], m = mi*16 + l15
        const float* hrow = hcur + (mi * 16 + l15) * H_;
        v2f a[8];
        #pragma unroll
        for (int k0 = 0; k0 < 8; ++k0) {
            a[k0].x = hrow[4 * k0 + khalf];
            a[k0].y = hrow[4 * k0 + khalf + 1];
        }
        // C tile = I_t
        const float* Irow = Ibuf + t * (B_ * H_);
        v8f acc;
        #pragma unroll
        for (int r = 0; r < 8; ++r)
            acc[r] = Irow[(mi * 16 + r + rowoff) * H_ + (ni * 16 + l15)];
        #pragma unroll
        for (int k0 = 0; k0 < 8; ++k0)
            acc = __builtin_amdgcn_wmma_f32_16x16x4_f32(
                false, a[k0], false, bb[k0], (short)0, acc, false, false);
        #pragma unroll
        for (int r = 0; r < 8; ++r)
            acc[r] = tanh_hw(acc[r]);

        __syncthreads();   // everyone done reading old h
        #pragma unroll
        for (int r = 0; r < 8; ++r) {
            int b  = mi * 16 + r + rowoff;
            int hh = ni * 16 + l15;
            hcur[b * H_ + hh] = acc[r];
            Hall[t * (B_ * H_) + b * H_ + hh] = acc[r];
        }
        __syncthreads();   // new h visible
    }
}

// ---------------------------------------------------------------------------
// Kernel 3a: stats pass.  Grid (t=64, q=8), 512 thr = 16 waves per block.
// Wave (mi, nj): mi = batch 16-row tile, nj = 16-col stripe of a 128-wide
// chunk; block q takes chunks vc = q, q+8, ... (interleaved stripes).
// WMMA GEMM + online (max, sumexp, argmax); per-block partial stats -> ws.
// ---------------------------------------------------------------------------
__global__ __launch_bounds__(512) void stats_kernel(
    const float* __restrict__ Hall, const float* __restrict__ Wo,
    const float* __restrict__ bo,
    float* __restrict__ sm_ws, float* __restrict__ ss_ws,
    float* __restrict__ sb_ws, int* __restrict__ si_ws)
{
    int t = blockIdx.x;
    int q = blockIdx.y;
    int tid  = threadIdx.x;
    int w    = tid >> 5;
    int lane = tid & 31;
    int mi = w >> 3;       // 0..1
    int nj = w & 7;        // 0..7
    int l15    = lane & 15;
    int khalf  = (lane < 16) ? 0 : 2;
    int rowoff = (lane < 16) ? 0 : 8;

    // A regs (h_t tile)
    const float* hrow = Hall + t * (B_ * H_) + (mi * 16 + l15) * H_;
    v2f a[8];
    #pragma unroll
    for (int k0 = 0; k0 < 8; ++k0) {
        a[k0].x = hrow[4 * k0 + khalf];
        a[k0].y = hrow[4 * k0 + khalf + 1];
    }

    float m_run[8], s_run[8], best[8];
    int   bidx[8];
    #pragma unroll
    for (int r = 0; r < 8; ++r) {
        m_run[r] = -3.0e38f; s_run[r] = 0.0f; best[r] = -3.0e38f; bidx[r] = 0;
    }

    for (int vc = q; vc < NCHUNK_; vc += Q_) {
        int n0 = vc * 128 + nj * 16;
        int v  = n0 + l15;
        float bov = bo[v];
        v8f acc;
        #pragma unroll
        for (int r = 0; r < 8; ++r) acc[r] = bov;
        const float* wrow = Wo + (size_t)v * H_;
        #pragma unroll
        for (int k0 = 0; k0 < 8; ++k0) {
            v2f bmat;
            bmat.x = wrow[4 * k0 + khalf];
            bmat.y = wrow[4 * k0 + khalf + 1];
            acc = __builtin_amdgcn_wmma_f32_16x16x4_f32(
                false, a[k0], false, bmat, (short)0, acc, false, false);
        }
        #pragma unroll
        for (int r = 0; r < 8; ++r) {
            // row-wise reduce over the 16 columns held by this half-wave
            float vmax = acc[r]; int imax = v;
            #pragma unroll
            for (int sft = 1; sft < 16; sft <<= 1) {
                float ov = __shfl_xor(vmax, sft, 32);
                int   oi = __shfl_xor(imax, sft, 32);
                if (ov > vmax || (ov == vmax && oi < imax)) { vmax = ov; imax = oi; }
            }
            float e = __expf(acc[r] - vmax);
            #pragma unroll
            for (int sft = 1; sft < 16; sft <<= 1)
                e += __shfl_xor(e, sft, 32);
            float nm = fmaxf(m_run[r], vmax);
            s_run[r] = s_run[r] * __expf(m_run[r] - nm) + e * __expf(vmax - nm);
            m_run[r] = nm;
            if (vmax > best[r]) { best[r] = vmax; bidx[r] = imax; }  // chunks ascend in v
        }
    }

    // Merge across the 8 nj waves in LDS, then write per-(t,q) partials.
    __shared__ float sm[8][32], ss[8][32], sb[8][32];
    __shared__ int   si[8][32];
    if (l15 == 0) {   // lanes 0 and 16 carry rows r and r+8
        #pragma unroll
        for (int r = 0; r < 8; ++r) {
            int row = mi * 16 + r + rowoff;
            sm[nj][row] = m_run[r]; ss[nj][row] = s_run[r];
            sb[nj][row] = best[r];  si[nj][row] = bidx[r];
        }
    }
    __syncthreads();
    if (tid < 32) {
        int row = tid;
        float m = -3.0e38f, s = 0.0f, bv = -3.0e38f; int bix = 0;
        #pragma unroll
        for (int j = 0; j < 8; ++j) {
            float mj = sm[j][row], sj = ss[j][row];
            float nm = fmaxf(m, mj);
            s = s * __expf(m - nm) + sj * __expf(mj - nm);
            m = nm;
            float bj = sb[j][row]; int ij = si[j][row];
            if (bj > bv || (bj == bv && ij < bix)) { bv = bj; bix = ij; }
        }
        int pidx = (t * Q_ + q) * 32 + row;
        sm_ws[pidx] = m; ss_ws[pidx] = s; sb_ws[pidx] = bv; si_ws[pidx] = bix;
    }
}

// ---------------------------------------------------------------------------
// Kernel 3b: write pass.  Grid (t=64, q=8) = 512 blocks so the 262 MB
// streaming write spans many WGPs (write bandwidth is the roofline floor).
// Each block redundantly merges the 8 partial stats (trivial), recomputes
// its GEMM stripe, and writes logp = x - (m + log s) with non-temporal
// stores so the stream does not evict Wo/bo from L2.
// ---------------------------------------------------------------------------
__global__ __launch_bounds__(512) void write_kernel(
    const float* __restrict__ Hall, const float* __restrict__ Wo,
    const float* __restrict__ bo,
    const float* __restrict__ sm_ws, const float* __restrict__ ss_ws,
    const float* __restrict__ sb_ws, const int* __restrict__ si_ws,
    float* __restrict__ out)
{
    int t = blockIdx.x;
    int q = blockIdx.y;
    int tid  = threadIdx.x;
    int w    = tid >> 5;
    int lane = tid & 31;
    int mi = w >> 3;
    int nj = w & 7;
    int l15    = lane & 15;
    int khalf  = (lane < 16) ? 0 : 2;
    int rowoff = (lane < 16) ? 0 : 8;

    __shared__ float snorm[32];
    if (tid < 32) {
        int row = tid;
        float m = -3.0e38f, s = 0.0f, bv = -3.0e38f; int bix = 0;
        #pragma unroll
        for (int j = 0; j < Q_; ++j) {
            int pidx = (t * Q_ + j) * 32 + row;
            float mj = sm_ws[pidx], sj = ss_ws[pidx];
            float nm = fmaxf(m, mj);
            s = s * __expf(m - nm) + sj * __expf(mj - nm);
            m = nm;
            float bj = sb_ws[pidx]; int ij = si_ws[pidx];
            if (bj > bv || (bj == bv && ij < bix)) { bv = bj; bix = ij; }
        }
        snorm[row] = m + __logf(s);
        if (q == 0)
            out[(size_t)B_ * T_ * V_ + row * T_ + t] = (float)bix;   // preds
    }
    __syncthreads();

    // A regs (h_t tile)
    const float* hrow = Hall + t * (B_ * H_) + (mi * 16 + l15) * H_;
    v2f a[8];
    #pragma unroll
    for (int k0 = 0; k0 < 8; ++k0) {
        a[k0].x = hrow[4 * k0 + khalf];
        a[k0].y = hrow[4 * k0 + khalf + 1];
    }
    float norm[8];
    #pragma unroll
    for (int r = 0; r < 8; ++r) norm[r] = snorm[mi * 16 + r + rowoff];

    for (int vc = q; vc < NCHUNK_; vc += Q_) {
        int n0 = vc * 128 + nj * 16;
        int v  = n0 + l15;
        float bov = bo[v];
        v8f acc;
        #pragma unroll
        for (int r = 0; r < 8; ++r) acc[r] = bov;
        const float* wrow = Wo + (size_t)v * H_;
        #pragma unroll
        for (int k0 = 0; k0 < 8; ++k0) {
            v2f bmat;
            bmat.x = wrow[4 * k0 + khalf];
            bmat.y = wrow[4 * k0 + khalf + 1];
            acc = __builtin_amdgcn_wmma_f32_16x16x4_f32(
                false, a[k0], false, bmat, (short)0, acc, false, false);
        }
        #pragma unroll
        for (int r = 0; r < 8; ++r) {
            int row = mi * 16 + r + rowoff;
            __builtin_nontemporal_store(acc[r] - norm[r],
                                        &out[((size_t)row * T_ + t) * V_ + v]);
        }
    }
}

// ---------------------------------------------------------------------------
extern "C" void kernel_launch(void* const* d_in, const int* in_sizes, int n_in,
                              void* d_out, int out_size, void* d_ws, size_t ws_size,
                              hipStream_t stream) {
    const int*   y    = (const int*)  d_in[0];
    const float* enc  = (const float*)d_in[1];
    const float* emb  = (const float*)d_in[2];
    const float* Wi   = (const float*)d_in[3];
    const float* bi   = (const float*)d_in[4];
    const float* Wh   = (const float*)d_in[5];
    const float* bh   = (const float*)d_in[6];
    const float* Wo   = (const float*)d_in[7];
    const float* bo   = (const float*)d_in[8];
    float* out  = (float*)d_out;

    // Workspace layout (floats): Ibuf | Hall | sm | ss | sb | si  (~768 KB)
    float* Ibuf  = (float*)d_ws;                 // T*B*H
    float* Hall  = Ibuf  + T_ * B_ * H_;         // T*B*H
    float* sm_ws = Hall  + T_ * B_ * H_;         // T*Q*32
    float* ss_ws = sm_ws + T_ * Q_ * 32;
    float* sb_ws = ss_ws + T_ * Q_ * 32;
    int*   si_ws = (int*)(sb_ws + T_ * Q_ * 32);

    hipLaunchKernelGGL(i2h_kernel, dim3((T_ * B_) / 8), dim3(256), 0, stream,
                       y, emb, Wi, bi, bh, Ibuf);
    hipLaunchKernelGGL(recurrent_kernel, dim3(1), dim3(128), 0, stream,
                       enc, Wh, Ibuf, Hall);
    hipLaunchKernelGGL(stats_kernel, dim3(T_, Q_), dim3(512), 0, stream,
                       Hall, Wo, bo, sm_ws, ss_ws, sb_ws, si_ws);
    hipLaunchKernelGGL(write_kernel, dim3(T_, Q_), dim3(512), 0, stream,
                       Hall, Wo, bo, sm_ws, ss_ws, sb_ws, si_ws, out);
}